// SyntacticDistanceGate_62680752718165
// MI455X (gfx1250) — compile-verified
//
#include <hip/hip_runtime.h>
#include <math.h>

#define BB 4
#define LL 2048
#define DD 768
#define HH 2
#define RR 3

typedef float v2f __attribute__((ext_vector_type(2)));
typedef float v8f __attribute__((ext_vector_type(8)));

__device__ __forceinline__ float gelu_exact(float x) {
    // jax.nn.gelu(approximate=False): 0.5*x*(1+erf(x/sqrt(2)))
    return 0.5f * x * (1.0f + erff(x * 0.70710678118654752440f));
}

// C[M=8192, N=768] = X[M, K=768] * W^T (W is [N, K] row-major), then bias + exact GELU.
// One wave per block computing a 32x64 block of C as 2x4 WMMA 16x16 tiles.
// Hand double-buffered K-loop: all fragment loads for k+4 and k+8 issue before the
// WMMA groups for k and k+4, so VMEM latency overlaps V_WMMA_F32_16X16X4_F32.
// The compiler's machine pipeliner is disabled: it otherwise duplicates the 64
// accumulator VGPRs across two register homes and injects 32 v_mov_b64 per
// iteration (seen in round-2 asm).
__global__ __launch_bounds__(32)
void gemm_bias_gelu_wmma(const float* __restrict__ X,
                         const float* __restrict__ W,
                         const float* __restrict__ bias,
                         float* __restrict__ Y)
{
    const int lane = threadIdx.x & 31;
    const int half = lane >> 4;   // selects K pair {0,1} vs {2,3}
    const int l16  = lane & 15;   // M index for A-frag, N index for B-frag / C
    const int o0 = blockIdx.x * 64;
    const int m0 = blockIdx.y * 32;

    // Divergent 32-bit element offsets; uniform pointer advances in SGPRs.
    int offx[2], offw[4];
    #pragma unroll
    for (int mt = 0; mt < 2; ++mt)
        offx[mt] = (m0 + mt * 16 + l16) * DD + 2 * half;
    #pragma unroll
    for (int nt = 0; nt < 4; ++nt)
        offw[nt] = (o0 + nt * 16 + l16) * DD + 2 * half;

    v8f acc[2][4];
    #pragma unroll
    for (int mt = 0; mt < 2; ++mt)
        #pragma unroll
        for (int nt = 0; nt < 4; ++nt)
            acc[mt][nt] = (v8f)0.0f;

    v2f a0[2], b0[4];
    const float* Xk = X;
    const float* Wk = W;

    // Prologue: fragments for k=0.
    #pragma unroll
    for (int mt = 0; mt < 2; ++mt) a0[mt] = *(const v2f*)(Xk + offx[mt]);
    #pragma unroll
    for (int nt = 0; nt < 4; ++nt) b0[nt] = *(const v2f*)(Wk + offw[nt]);

    // Main loop: iteration it handles k0 = 8*it and 8*it+4; last pair peeled.
    #pragma clang loop unroll(disable) pipeline(disable)
    for (int it = 0; it < DD / 8 - 1; ++it) {
        v2f a1[2], b1[4], an[2], bn[4];
        // Issue ALL loads for this iteration's second group (k+4) and the next
        // iteration's first group (k+8) before any WMMA touches the pipe.
        #pragma unroll
        for (int mt = 0; mt < 2; ++mt) a1[mt] = *(const v2f*)(Xk + offx[mt] + 4);
        #pragma unroll
        for (int nt = 0; nt < 4; ++nt) b1[nt] = *(const v2f*)(Wk + offw[nt] + 4);
        #pragma unroll
        for (int mt = 0; mt < 2; ++mt) an[mt] = *(const v2f*)(Xk + offx[mt] + 8);
        #pragma unroll
        for (int nt = 0; nt < 4; ++nt) bn[nt] = *(const v2f*)(Wk + offw[nt] + 8);

        #pragma unroll
        for (int mt = 0; mt < 2; ++mt)
            #pragma unroll
            for (int nt = 0; nt < 4; ++nt)
                acc[mt][nt] = __builtin_amdgcn_wmma_f32_16x16x4_f32(
                    false, a0[mt], false, b0[nt], (short)0, acc[mt][nt], false, false);
        #pragma unroll
        for (int mt = 0; mt < 2; ++mt)
            #pragma unroll
            for (int nt = 0; nt < 4; ++nt)
                acc[mt][nt] = __builtin_amdgcn_wmma_f32_16x16x4_f32(
                    false, a1[mt], false, b1[nt], (short)0, acc[mt][nt], false, false);

        // Rotate buffers (single def/use; coalesced by regalloc, no real copies).
        #pragma unroll
        for (int mt = 0; mt < 2; ++mt) a0[mt] = an[mt];
        #pragma unroll
        for (int nt = 0; nt < 4; ++nt) b0[nt] = bn[nt];
        Xk += 8; Wk += 8;
    }

    // Peel: a0/b0 hold k = DD-8; load and process k = DD-4.
    {
        v2f a1[2], b1[4];
        #pragma unroll
        for (int mt = 0; mt < 2; ++mt) a1[mt] = *(const v2f*)(Xk + offx[mt] + 4);
        #pragma unroll
        for (int nt = 0; nt < 4; ++nt) b1[nt] = *(const v2f*)(Wk + offw[nt] + 4);
        #pragma unroll
        for (int mt = 0; mt < 2; ++mt)
            #pragma unroll
            for (int nt = 0; nt < 4; ++nt)
                acc[mt][nt] = __builtin_amdgcn_wmma_f32_16x16x4_f32(
                    false, a0[mt], false, b0[nt], (short)0, acc[mt][nt], false, false);
        #pragma unroll
        for (int mt = 0; mt < 2; ++mt)
            #pragma unroll
            for (int nt = 0; nt < 4; ++nt)
                acc[mt][nt] = __builtin_amdgcn_wmma_f32_16x16x4_f32(
                    false, a1[mt], false, b1[nt], (short)0, acc[mt][nt], false, false);
    }

    // Epilogue: C[v] @ lane -> row = v + 8*half, col = l16
    #pragma unroll
    for (int nt = 0; nt < 4; ++nt) {
        const int col = o0 + nt * 16 + l16;
        const float bv = bias[col];
        #pragma unroll
        for (int mt = 0; mt < 2; ++mt) {
            #pragma unroll
            for (int v = 0; v < 8; ++v) {
                const int rowm = m0 + mt * 16 + v + 8 * half;
                Y[(size_t)rowm * DD + col] = gelu_exact(acc[mt][nt][v] + bv);
            }
        }
    }
}

// d[b,h,l] = tanh(b3[h] + sum_{r=0..2} sum_d w3[h,d,r] * h2[b, l+r-2, d])
// Also writes dist_out[b,l] = mean over heads.
__global__ __launch_bounds__(256)
void conv_tanh_kernel(const float* __restrict__ h2,
                      const float* __restrict__ w3,
                      const float* __restrict__ b3,
                      float* __restrict__ dbuf,
                      float* __restrict__ dist_out)
{
    const int bl = blockIdx.x;
    const int b = bl / LL;
    const int l = bl % LL;
    const int t = threadIdx.x;

    float s0 = 0.0f, s1 = 0.0f;
    #pragma unroll
    for (int r = 0; r < RR; ++r) {
        const int ls = l + r - (RR - 1);   // l + r - 2 (zero-padded below 0)
        if (ls < 0) continue;
        const float* row = h2 + ((size_t)b * LL + ls) * DD;
        for (int d = t; d < DD; d += 256) {
            const float v = row[d];
            s0 += v * w3[(0 * DD + d) * RR + r];
            s1 += v * w3[(1 * DD + d) * RR + r];
        }
    }
    __shared__ float red0[256];
    __shared__ float red1[256];
    red0[t] = s0; red1[t] = s1;
    __syncthreads();
    for (int s = 128; s > 0; s >>= 1) {
        if (t < s) { red0[t] += red0[t + s]; red1[t] += red1[t + s]; }
        __syncthreads();
    }
    if (t == 0) {
        const float d0 = tanhf(red0[0] + b3[0]);
        const float d1 = tanhf(red1[0] + b3[1]);
        dbuf[((size_t)b * HH + 0) * LL + l] = d0;
        dbuf[((size_t)b * HH + 1) * LL + l] = d1;
        dist_out[(size_t)b * LL + l] = 0.5f * (d0 + d1);
    }
}

// gate[g, i, j]: suffix cumprod of masked alpha (j<i) times prefix cumprod (j>i).
// One 256-thread block per row; 8 contiguous j per thread; block-wide product scans.
__global__ __launch_bounds__(256)
void gate_kernel(const float* __restrict__ dbuf, float* __restrict__ gate)
{
    const int rowid = blockIdx.x;          // g*L + i
    const int g = rowid >> 11;
    const int i = rowid & (LL - 1);
    const int t = threadIdx.x;
    const int j0 = t * 8;

    const float* dr = dbuf + (size_t)g * LL;
    const float di = dr[i];

    const float4 p0 = *(const float4*)(dr + j0);
    const float4 p1 = *(const float4*)(dr + j0 + 4);
    const float dj[8] = {p0.x, p0.y, p0.z, p0.w, p1.x, p1.y, p1.z, p1.w};
    float alpha[8];
    #pragma unroll
    for (int k = 0; k < 8; ++k) {
        float v = di - dj[k];
        v = fminf(1.0f, fmaxf(-1.0f, v));
        alpha[k] = (v + 1.0f) * 0.5f;
    }

    __shared__ float s[256];

    // upper: inclusive prefix product of (j > i ? alpha : 1)
    float up[8];
    {
        float run = 1.0f;
        #pragma unroll
        for (int k = 0; k < 8; ++k) {
            const float v = (j0 + k > i) ? alpha[k] : 1.0f;
            run *= v;
            up[k] = run;
        }
        s[t] = run;
        __syncthreads();
        for (int off = 1; off < 256; off <<= 1) {
            const float v = (t >= off) ? s[t - off] : 1.0f;
            __syncthreads();
            s[t] *= v;
            __syncthreads();
        }
        const float pre = (t == 0) ? 1.0f : s[t - 1];
        #pragma unroll
        for (int k = 0; k < 8; ++k) up[k] *= pre;
    }
    __syncthreads();

    // lower: inclusive suffix product of (j < i ? alpha : 1)
    float lo[8];
    {
        float run = 1.0f;
        #pragma unroll
        for (int k = 7; k >= 0; --k) {
            const float v = (j0 + k < i) ? alpha[k] : 1.0f;
            run *= v;
            lo[k] = run;
        }
        s[t] = run;
        __syncthreads();
        for (int off = 1; off < 256; off <<= 1) {
            const float v = (t + off < 256) ? s[t + off] : 1.0f;
            __syncthreads();
            s[t] *= v;
            __syncthreads();
        }
        const float post = (t == 255) ? 1.0f : s[t + 1];
        #pragma unroll
        for (int k = 0; k < 8; ++k) lo[k] *= post;
    }

    float4 o0, o1;
    o0.x = lo[0] * up[0]; o0.y = lo[1] * up[1];
    o0.z = lo[2] * up[2]; o0.w = lo[3] * up[3];
    o1.x = lo[4] * up[4]; o1.y = lo[5] * up[5];
    o1.z = lo[6] * up[6]; o1.w = lo[7] * up[7];
    float* dst = gate + (size_t)rowid * LL + j0;
    *(float4*)dst = o0;
    *(float4*)(dst + 4) = o1;
}

extern "C" void kernel_launch(void* const* d_in, const int* in_sizes, int n_in,
                              void* d_out, int out_size, void* d_ws, size_t ws_size,
                              hipStream_t stream) {
    const float* x  = (const float*)d_in[0];
    const float* w1 = (const float*)d_in[1];
    const float* b1 = (const float*)d_in[2];
    const float* w2 = (const float*)d_in[3];
    const float* b2 = (const float*)d_in[4];
    const float* w3 = (const float*)d_in[5];
    const float* b3 = (const float*)d_in[6];

    float* out = (float*)d_out;
    const size_t GATE = (size_t)BB * HH * LL * LL;   // 33,554,432 floats
    float* gate = out;
    float* dist = out + GATE;                        // [B, L, 1] tail

    // Scratch for h1/h2 lives inside the gate region of d_out (12.6M < 33.5M
    // floats) and is fully overwritten by gate_kernel afterwards -> d_ws only
    // needs B*H*L floats (64 KB) for the distance vector.
    float* h1 = out;
    float* h2 = out + (size_t)BB * LL * DD;
    float* dbuf = (float*)d_ws;

    dim3 ggrid(DD / 64, (BB * LL) / 32);             // 12 x 256 blocks, 1 wave each
    gemm_bias_gelu_wmma<<<ggrid, 32, 0, stream>>>(x,  w1, b1, h1);
    gemm_bias_gelu_wmma<<<ggrid, 32, 0, stream>>>(h1, w2, b2, h2);
    conv_tanh_kernel<<<BB * LL, 256, 0, stream>>>(h2, w3, b3, dbuf, dist);
    gate_kernel<<<BB * HH * LL, 256, 0, stream>>>(dbuf, gate);
}